// GraphEncoder_53317724013093
// MI455X (gfx1250) — compile-verified
//
#include <hip/hip_runtime.h>
#include <hip/hip_bf16.h>
#include <math.h>

// Problem constants (match reference setup_inputs)
#define NN 16384     // total nodes
#define EE 131072    // edges
#define FD 64        // node feature dim
#define HD 4         // heads
#define CD 64        // per-head dim
#define HFD 256      // H*C
#define NB 16        // graphs
#define NG 1024      // nodes per graph
#define EF 16        // edge feature dim

typedef __attribute__((ext_vector_type(2))) float v2f;
typedef __attribute__((ext_vector_type(8))) float v8f;

// ---------- order-preserving float<->uint map for atomic max ----------
__device__ __forceinline__ unsigned int fenc(float f) {
  unsigned int u = __float_as_uint(f);
  return (u & 0x80000000u) ? ~u : (u | 0x80000000u);
}
__device__ __forceinline__ float fdec(unsigned int u) {
  return (u & 0x80000000u) ? __uint_as_float(u & 0x7fffffffu) : __uint_as_float(~u);
}

// ---------- fp32 WMMA GEMM: C = A(MxK) @ W(KxNc) + bias ----------
// One wave per 16x16 output tile, v_wmma_f32_16x16x4_f32, K%4==0, M%128==0, Nc%16==0.
__global__ void wmma_gemm_bias_kernel(const float* __restrict__ A,
                                      const float* __restrict__ W,
                                      const float* __restrict__ bias,
                                      float* __restrict__ C,
                                      int M, int K, int Nc) {
  const int lane = threadIdx.x & 31;
  const int half = lane >> 4;   // 0: lanes 0-15, 1: lanes 16-31
  const int l16  = lane & 15;
  const int n0 = blockIdx.x * 16;
  const int m0 = (blockIdx.y * blockDim.y + threadIdx.y) * 16;

  v8f acc = {};
  const float* Abase = A + (size_t)(m0 + l16) * K;
  for (int k0 = 0; k0 < K; k0 += 4) {
    // A 16x4: lane row = l16; vgpr0 = K(k0+2*half), vgpr1 = K(k0+1+2*half)
    v2f a = *(const v2f*)(Abase + k0 + 2 * half);
    // B 4x16: lane col = l16; vgpr0 = row k0+2*half, vgpr1 = row k0+1+2*half
    const float* wp = W + (size_t)(k0 + 2 * half) * Nc + n0 + l16;
    v2f b; b.x = wp[0]; b.y = wp[Nc];
    acc = __builtin_amdgcn_wmma_f32_16x16x4_f32(false, a, false, b, (short)0, acc,
                                                false, false);
  }
  const float bn = bias ? bias[n0 + l16] : 0.0f;
  float* Cp = C + (size_t)(m0 + 8 * half) * Nc + n0 + l16;
#pragma unroll
  for (int r = 0; r < 8; ++r) Cp[(size_t)r * Nc] = acc[r] + bn;
}

// ---------- init: xcur = x, emask = 1, nmask = 1 ----------
__global__ void init_kernel(const float* __restrict__ x, float* __restrict__ xcur,
                            unsigned char* __restrict__ emask,
                            unsigned char* __restrict__ nmask) {
  int i = blockIdx.x * blockDim.x + threadIdx.x;
  if (i < NN * FD) xcur[i] = x[i];
  if (i < EE) emask[i] = 1;
  if (i < NN) nmask[i] = 1;
}

// ---------- per-layer reset of aggregation buffers ----------
__global__ void conv_init_kernel(float* __restrict__ outb,
                                 unsigned int* __restrict__ mmax,
                                 float* __restrict__ ssum) {
  int i = blockIdx.x * blockDim.x + threadIdx.x;
  if (i < NN * HFD) outb[i] = 0.0f;
  if (i < NN * HD) { mmax[i] = 0u; ssum[i] = 0.0f; }   // enc(min) == 0
}

// ---------- pass A: logits + segment max (one wave per edge) ----------
__global__ void edge_logits_kernel(const float* __restrict__ q, const float* __restrict__ k,
                                   const float* __restrict__ ea, const float* __restrict__ We,
                                   const float* __restrict__ be,
                                   const int* __restrict__ src, const int* __restrict__ dst,
                                   const unsigned char* __restrict__ emask,
                                   float* __restrict__ logit,
                                   unsigned int* __restrict__ mmax) {
  int wid  = blockIdx.x * 8 + (threadIdx.x >> 5);
  int lane = threadIdx.x & 31;
  int s = src[wid], d = dst[wid];
  float ef[EF];
  const float* eap = ea + (size_t)wid * EF;
#pragma unroll
  for (int fi = 0; fi < EF; ++fi) ef[fi] = eap[fi];
  int c0 = lane * 8;
  const float* qd = q + (size_t)d * HFD + c0;
  const float* ks = k + (size_t)s * HFD + c0;
  float part = 0.0f;
#pragma unroll
  for (int j = 0; j < 8; ++j) {
    int c = c0 + j;
    float e = be[c];
#pragma unroll
    for (int fi = 0; fi < EF; ++fi) e += ef[fi] * We[fi * HFD + c];
    part += qd[j] * (ks[j] + e);
  }
  part += __shfl_xor(part, 1);
  part += __shfl_xor(part, 2);
  part += __shfl_xor(part, 4);
  if ((lane & 7) == 0) {
    int h = lane >> 3;
    float lg = part * 0.125f;             // 1/sqrt(C)
    if (!emask[wid]) lg = -1e9f;
    logit[(size_t)wid * HD + h] = lg;
    atomicMax(&mmax[(size_t)d * HD + h], fenc(lg));
  }
}

// ---------- pass B: w = exp(logit-m)*emask, segment sum ----------
__global__ void edge_weights_kernel(const int* __restrict__ dst,
                                    const unsigned char* __restrict__ emask,
                                    const unsigned int* __restrict__ mmax,
                                    float* __restrict__ wbuf, float* __restrict__ ssum) {
  int i = blockIdx.x * blockDim.x + threadIdx.x;  // EE*HD
  int e = i >> 2, h = i & 3;
  float w = 0.0f;
  if (emask[e]) {
    int d = dst[e];
    w = expf(wbuf[i] - fdec(mmax[(size_t)d * HD + h]));
    atomicAdd(&ssum[(size_t)d * HD + h], w);
  }
  wbuf[i] = w;
}

// ---------- pass C: msg = (v[src]+e)*alpha, scatter add (one wave per edge) ----------
__global__ void edge_message_kernel(const float* __restrict__ v, const float* __restrict__ ea,
                                    const float* __restrict__ We, const float* __restrict__ be,
                                    const int* __restrict__ src, const int* __restrict__ dst,
                                    const unsigned char* __restrict__ emask,
                                    const float* __restrict__ wbuf,
                                    const float* __restrict__ ssum,
                                    float* __restrict__ outb) {
  int wid = blockIdx.x * 8 + (threadIdx.x >> 5);
  if (!emask[wid]) return;                 // uniform per wave
  int lane = threadIdx.x & 31;
  int s = src[wid], d = dst[wid];
  int h = lane >> 3;
  float alpha = wbuf[(size_t)wid * HD + h] / (ssum[(size_t)d * HD + h] + 1e-16f);
  float ef[EF];
  const float* eap = ea + (size_t)wid * EF;
#pragma unroll
  for (int fi = 0; fi < EF; ++fi) ef[fi] = eap[fi];
  int c0 = lane * 8;
  const float* vs = v + (size_t)s * HFD + c0;
  float* op = outb + (size_t)d * HFD + c0;
#pragma unroll
  for (int j = 0; j < 8; ++j) {
    int c = c0 + j;
    float e = be[c];
#pragma unroll
    for (int fi = 0; fi < EF; ++fi) e += ef[fi] * We[fi * HFD + c];
    atomicAdd(&op[j], (vs[j] + e) * alpha);
  }
}

// ---------- beta gate: outb = beta*xr + (1-beta)*out (one wave per node) ----------
__global__ void beta_combine_kernel(const float* __restrict__ xr,
                                    const float* __restrict__ Wb,
                                    float* __restrict__ outb) {
  int wid  = blockIdx.x * 8 + (threadIdx.x >> 5);   // node
  int lane = threadIdx.x & 31;
  int c0 = lane * 8;
  const float* xp = xr + (size_t)wid * HFD + c0;
  float* op = outb + (size_t)wid * HFD + c0;
  float o[8], r[8];
  float part = 0.0f;
#pragma unroll
  for (int j = 0; j < 8; ++j) {
    o[j] = op[j]; r[j] = xp[j];
    int c = c0 + j;
    part += o[j] * Wb[c] + r[j] * Wb[HFD + c] + (o[j] - r[j]) * Wb[2 * HFD + c];
  }
  part += __shfl_xor(part, 1);
  part += __shfl_xor(part, 2);
  part += __shfl_xor(part, 4);
  part += __shfl_xor(part, 8);
  part += __shfl_xor(part, 16);
  float beta = 1.0f / (1.0f + expf(-part));
#pragma unroll
  for (int j = 0; j < 8; ++j) op[j] = beta * r[j] + (1.0f - beta) * o[j];
}

// ---------- relu + BN(eval) + residual ----------
__global__ void post_kernel(const float* __restrict__ h, const float* __restrict__ gamma,
                            const float* __restrict__ bb, float* __restrict__ xcur) {
  int i = blockIdx.x * blockDim.x + threadIdx.x;  // NN*FD
  int c = i & (FD - 1);
  float t = fmaxf(h[i], 0.0f) * 0.99999500003749969f * gamma[c] + bb[c];
  xcur[i] = t + xcur[i];
}

// ---------- pooling: scores ----------
__global__ void score_kernel(const float* __restrict__ xcur, const float* __restrict__ w,
                             const unsigned char* __restrict__ nmask,
                             float* __restrict__ score, float* __restrict__ smask) {
  int n = blockIdx.x * blockDim.x + threadIdx.x;
  if (n >= NN) return;
  const float* xp = xcur + (size_t)n * FD;
  float num = 0.0f, wn = 0.0f;
#pragma unroll
  for (int f = 0; f < FD; ++f) { float wf = w[f]; num += xp[f] * wf; wn += wf * wf; }
  float sc = tanhf(num * rsqrtf(wn));
  score[n] = sc;
  smask[n] = nmask[n] ? sc : -2.0f;
}

// ---------- pooling: per-graph k-th largest via LDS bitonic sort ----------
__global__ void topk_thresh_kernel(const float* __restrict__ smask,
                                   float* __restrict__ thresh, int kk) {
  __shared__ float sd[NG];
  int g = blockIdx.x, t = threadIdx.x;
  sd[t] = smask[g * NG + t];
  __syncthreads();
  for (int k = 2; k <= NG; k <<= 1) {
    for (int j = k >> 1; j > 0; j >>= 1) {
      int ixj = t ^ j;
      if (ixj > t) {
        float a = sd[t], b = sd[ixj];
        bool up = ((t & k) == 0);
        if ((a > b) == up) { sd[t] = b; sd[ixj] = a; }   // ascending overall
      }
      __syncthreads();
    }
  }
  if (t == 0) thresh[g] = sd[NG - kk];
}

// ---------- pooling: apply mask/scale ----------
__global__ void apply_pool_kernel(const float* __restrict__ score,
                                  const float* __restrict__ smask,
                                  const float* __restrict__ thresh,
                                  float* __restrict__ xcur,
                                  unsigned char* __restrict__ nmask) {
  int n = blockIdx.x * blockDim.x + threadIdx.x;
  if (n >= NN) return;
  int g = n >> 10;
  bool nm = smask[n] >= thresh[g];
  float scale = nm ? score[n] : 0.0f;
  float* xp = xcur + (size_t)n * FD;
#pragma unroll
  for (int f = 0; f < FD; ++f) xp[f] *= scale;
  nmask[n] = nm ? 1 : 0;
}

__global__ void emask_update_kernel(const int* __restrict__ src, const int* __restrict__ dst,
                                    const unsigned char* __restrict__ nmask,
                                    unsigned char* __restrict__ emask) {
  int e = blockIdx.x * blockDim.x + threadIdx.x;
  if (e >= EE) return;
  emask[e] = (emask[e] && nmask[src[e]] && nmask[dst[e]]) ? 1 : 0;
}

// ---------- global max/mean pooling ----------
__global__ void global_rep_kernel(const float* __restrict__ xcur,
                                  const unsigned char* __restrict__ nmask,
                                  float* __restrict__ rep, float inv_kk) {
  int g = blockIdx.x, f = threadIdx.x;   // 64 threads
  float mx = -INFINITY, sm = 0.0f;
  const float* base = xcur + (size_t)g * NG * FD + f;
  const unsigned char* nm = nmask + (size_t)g * NG;
  for (int i = 0; i < NG; ++i) {
    float v = base[(size_t)i * FD];
    if (nm[i]) mx = fmaxf(mx, v);
    sm += v;                             // dead nodes already zeroed
  }
  rep[g * 2 * FD + f] = mx;
  rep[g * 2 * FD + FD + f] = sm * inv_kk;
}

__global__ void add_out_kernel(const float* __restrict__ r0, const float* __restrict__ r1,
                               float* __restrict__ out) {
  int i = blockIdx.x * blockDim.x + threadIdx.x;
  if (i < NB * 2 * FD) out[i] = r0[i] + r1[i];
}

extern "C" void kernel_launch(void* const* d_in, const int* in_sizes, int n_in,
                              void* d_out, int out_size, void* d_ws, size_t ws_size,
                              hipStream_t stream) {
  const float* x  = (const float*)d_in[0];
  const float* ea = (const float*)d_in[1];
  const int* ei   = (const int*)d_in[2];
  const int* srcp = ei;
  const int* dstp = ei + EE;
  auto P = [&](int i) { return (const float*)d_in[i]; };
  // param leaf layout (insertion order): 4 + 11*l for conv l, then transf, bn, pool_w
  // conv leaf order: Wq,bq,Wk,bk,Wv,bv,We,be,Ws,bs,Wb

  // workspace layout
  float* f = (float*)d_ws;
  float* qb     = f; f += (size_t)NN * HFD;
  float* kb     = f; f += (size_t)NN * HFD;
  float* vb     = f; f += (size_t)NN * HFD;
  float* xrb    = f; f += (size_t)NN * HFD;
  float* outb   = f; f += (size_t)NN * HFD;
  float* logitw = f; f += (size_t)EE * HD;
  unsigned int* mmax = (unsigned int*)f; f += (size_t)NN * HD;
  float* ssum   = f; f += (size_t)NN * HD;
  float* hbuf   = f; f += (size_t)NN * FD;
  float* xcur   = f; f += (size_t)NN * FD;
  float* score  = f; f += NN;
  float* smaskb = f; f += NN;
  float* thresh = f; f += NB;
  float* rep0   = f; f += NB * 2 * FD;
  float* rep1   = f; f += NB * 2 * FD;
  unsigned char* emaskb = (unsigned char*)f;
  unsigned char* nmaskb = emaskb + EE;

  auto gemm = [&](const float* A, const float* W, const float* b, float* C,
                  int M, int K, int Nc) {
    dim3 blk(32, 8);
    dim3 grd(Nc / 16, M / 128);
    wmma_gemm_bias_kernel<<<grd, blk, 0, stream>>>(A, W, b, C, M, K, Nc);
  };

  init_kernel<<<(NN * FD + 255) / 256, 256, 0, stream>>>(x, xcur, emaskb, nmaskb);

  auto run_block = [&](int l) {
    int cb = 4 + 11 * l;
    const float *Wq = P(cb + 0), *bq = P(cb + 1), *Wk = P(cb + 2), *bk = P(cb + 3);
    const float *Wv = P(cb + 4), *bv = P(cb + 5), *We = P(cb + 6), *be = P(cb + 7);
    const float *Ws = P(cb + 8), *bs = P(cb + 9), *Wb = P(cb + 10);
    const float *tW = P(48 + 2 * l), *tb = P(49 + 2 * l);
    const float *gm = P(56 + 2 * l), *bb = P(57 + 2 * l);

    conv_init_kernel<<<(NN * HFD + 255) / 256, 256, 0, stream>>>(outb, mmax, ssum);
    gemm(xcur, Wq, bq, qb,  NN, FD, HFD);
    gemm(xcur, Wk, bk, kb,  NN, FD, HFD);
    gemm(xcur, Wv, bv, vb,  NN, FD, HFD);
    gemm(xcur, Ws, bs, xrb, NN, FD, HFD);
    edge_logits_kernel<<<EE / 8, 256, 0, stream>>>(qb, kb, ea, We, be, srcp, dstp,
                                                   emaskb, logitw, mmax);
    edge_weights_kernel<<<EE * HD / 256, 256, 0, stream>>>(dstp, emaskb, mmax,
                                                           logitw, ssum);
    edge_message_kernel<<<EE / 8, 256, 0, stream>>>(vb, ea, We, be, srcp, dstp,
                                                    emaskb, logitw, ssum, outb);
    beta_combine_kernel<<<NN / 8, 256, 0, stream>>>(xrb, Wb, outb);
    gemm(outb, tW, tb, hbuf, NN, HFD, FD);
    post_kernel<<<NN * FD / 256, 256, 0, stream>>>(hbuf, gm, bb, xcur);
  };

  auto run_pool = [&](const float* pw, int kk, float* rep) {
    score_kernel<<<NN / 256, 256, 0, stream>>>(xcur, pw, nmaskb, score, smaskb);
    topk_thresh_kernel<<<NB, NG, 0, stream>>>(smaskb, thresh, kk);
    apply_pool_kernel<<<NN / 256, 256, 0, stream>>>(score, smaskb, thresh, xcur, nmaskb);
    emask_update_kernel<<<EE / 256, 256, 0, stream>>>(srcp, dstp, nmaskb, emaskb);
    global_rep_kernel<<<NB, FD, 0, stream>>>(xcur, nmaskb, rep, 1.0f / (float)kk);
  };

  run_block(0);
  run_block(1);
  run_pool(P(64), 512, rep0);   // pool_w[0], k=512
  run_block(2);
  run_block(3);
  run_pool(P(66), 256, rep1);   // pool_w[2], k=256
  add_out_kernel<<<(NB * 2 * FD + 255) / 256, 256, 0, stream>>>(rep0, rep1, (float*)d_out);
}